// GATEncoder_50431505989759
// MI455X (gfx1250) — compile-verified
//
#include <hip/hip_runtime.h>
#include <math.h>

typedef float v2f __attribute__((ext_vector_type(2)));
typedef float v8f __attribute__((ext_vector_type(8)));

#define NEG_SLOPE 0.2f

// ---------------------------------------------------------------------------
// WMMA GEMM: Hout[N x OUT] = X[N x K] @ W[K x OUT], fp32 via V_WMMA_F32_16X16X4_F32
// One wave (32 threads) computes one 16x16 output tile; K consumed 4 at a time.
// A-frag (16x4, ISA 7.12.2): lane<16 -> row=lane, {K=k0,k0+1}; lane>=16 -> {k0+2,k0+3}
// B-frag (4x16): lane<16 -> col=lane, {K=k0,k0+1}; lane>=16 -> {k0+2,k0+3}
// C/D: VGPR r, lane L -> D[r + 8*(L>=16)][L%16]
// ---------------------------------------------------------------------------
__global__ void __launch_bounds__(32)
gemm_wmma_kernel(const float* __restrict__ X, const float* __restrict__ W,
                 float* __restrict__ Hout, int n_rows, int K, int OUT) {
  const int lane = threadIdx.x;
  const int half = lane >> 4;
  const int l16  = lane & 15;
  const int row0 = blockIdx.x * 16;
  const int n0   = blockIdx.y * 16;

  const int arow = row0 + l16;
  const int arow_c = arow < n_rows ? arow : (n_rows - 1);
  const float* __restrict__ xrow = X + (size_t)arow_c * K;

  v8f acc = {};
  #pragma unroll 4
  for (int k0 = 0; k0 < K; k0 += 4) {
    const int ka = k0 + 2 * half;                 // even -> 8B aligned
    v2f a = *reinterpret_cast<const v2f*>(xrow + ka);
    v2f b;
    b.x = W[(size_t)ka * OUT + n0 + l16];
    b.y = W[(size_t)(ka + 1) * OUT + n0 + l16];
    acc = __builtin_amdgcn_wmma_f32_16x16x4_f32(false, a, false, b,
                                                (short)0, acc, false, false);
  }
  #pragma unroll
  for (int r = 0; r < 8; ++r) {
    const int row = row0 + r + 8 * half;
    if (row < n_rows) Hout[(size_t)row * OUT + n0 + l16] = acc[r];
  }
}

// ---------------------------------------------------------------------------
// Per-node attention logits: al_s[n,h] = <h[n,h,:], a_src[h,:]>, al_d likewise
// ---------------------------------------------------------------------------
__global__ void logits_kernel(const float* __restrict__ Hbuf,
                              const float* __restrict__ a_src,
                              const float* __restrict__ a_dst,
                              float* __restrict__ al_s, float* __restrict__ al_d,
                              int n_rows, int Hh, int C) {
  const int n = blockIdx.x * blockDim.x + threadIdx.x;
  if (n >= n_rows) return;
  const float* __restrict__ hrow = Hbuf + (size_t)n * Hh * C;
  for (int h = 0; h < Hh; ++h) {
    float s = 0.f, d = 0.f;
    #pragma unroll 8
    for (int c = 0; c < C; ++c) {
      const float v = hrow[h * C + c];
      s = fmaf(v, a_src[h * C + c], s);
      d = fmaf(v, a_dst[h * C + c], d);
    }
    al_s[(size_t)n * Hh + h] = s;
    al_d[(size_t)n * Hh + h] = d;
  }
}

__global__ void fill_kernel(float* __restrict__ p, float v, long long n) {
  const long long i = (long long)blockIdx.x * blockDim.x + threadIdx.x;
  if (i < n) p[i] = v;
}

__device__ __forceinline__ void atomic_max_float(float* addr, float val) {
  if (val >= 0.f) atomicMax((int*)addr, __float_as_int(val));
  else            atomicMin((unsigned int*)addr, __float_as_uint(val));
}

__device__ __forceinline__ void edge_nodes(const long long* __restrict__ ei,
                                           int E, long long e,
                                           long long& s, long long& d) {
  if (e < E) { s = ei[e]; d = ei[(size_t)E + e]; }
  else       { s = d = e - E; }                      // self loop
}

// Pass 1: segment max of leaky-relu logits over incoming edges per dst
__global__ void edge_max_kernel(const long long* __restrict__ ei, int E, int n_rows,
                                const float* __restrict__ al_s,
                                const float* __restrict__ al_d,
                                float* __restrict__ m, int Hh) {
  const long long e = (long long)blockIdx.x * blockDim.x + threadIdx.x;
  if (e >= (long long)E + n_rows) return;
  long long s, d; edge_nodes(ei, E, e, s, d);
  for (int h = 0; h < Hh; ++h) {
    float v = al_s[s * Hh + h] + al_d[d * Hh + h];
    v = v > 0.f ? v : NEG_SLOPE * v;
    atomic_max_float(&m[d * Hh + h], v);
  }
}

// Pass 2: ex = exp(e - m[dst]); z[dst] += ex  (ex cached for the aggregate pass)
__global__ void edge_expsum_kernel(const long long* __restrict__ ei, int E, int n_rows,
                                   const float* __restrict__ al_s,
                                   const float* __restrict__ al_d,
                                   const float* __restrict__ m,
                                   float* __restrict__ z, float* __restrict__ ex,
                                   int Hh) {
  const long long e = (long long)blockIdx.x * blockDim.x + threadIdx.x;
  if (e >= (long long)E + n_rows) return;
  long long s, d; edge_nodes(ei, E, e, s, d);
  for (int h = 0; h < Hh; ++h) {
    float v = al_s[s * Hh + h] + al_d[d * Hh + h];
    v = v > 0.f ? v : NEG_SLOPE * v;
    const float t = __expf(v - m[d * Hh + h]);
    ex[e * Hh + h] = t;
    atomicAdd(&z[d * Hh + h], t);
  }
}

// Pass 3: accum[dst, :] += (ex/z[dst]) * h[src, :]; (HC/4) threads per edge,
// each handles a float4 channel group -> 16B gather + 4 L2 atomic f32 adds.
__global__ void aggregate_kernel(const long long* __restrict__ ei, int E, int n_rows,
                                 const float* __restrict__ Hbuf,
                                 const float* __restrict__ ex,
                                 const float* __restrict__ z,
                                 float* __restrict__ accum,
                                 int Hh, int C, int gshift) {
  const long long tid = (long long)blockIdx.x * blockDim.x + threadIdx.x;
  const long long Etot = (long long)E + n_rows;
  if (tid >= (Etot << gshift)) return;
  const long long e = tid >> gshift;
  const int g = (int)(tid & ((1 << gshift) - 1));
  const int c4 = g * 4;
  const int h = c4 / 64;                              // C == 64
  const int HC = Hh * C;
  long long s, d; edge_nodes(ei, E, e, s, d);
  const float alpha = ex[e * Hh + h] / z[d * Hh + h];
  const float4 hv = *reinterpret_cast<const float4*>(Hbuf + s * HC + c4);
  float* outp = accum + d * HC + c4;
  atomicAdd(outp + 0, alpha * hv.x);
  atomicAdd(outp + 1, alpha * hv.y);
  atomicAdd(outp + 2, alpha * hv.z);
  atomicAdd(outp + 3, alpha * hv.w);
}

__global__ void bias_act_kernel(const float* __restrict__ accum,
                                const float* __restrict__ bias,
                                float* __restrict__ out,
                                long long total, int Fmask, int do_relu) {
  const long long i = (long long)blockIdx.x * blockDim.x + threadIdx.x;
  if (i >= total) return;
  float v = accum[i] + bias[(int)(i & Fmask)];
  if (do_relu) v = fmaxf(v, 0.f);
  out[i] = v;
}

// ---------------------------------------------------------------------------
extern "C" void kernel_launch(void* const* d_in, const int* in_sizes, int n_in,
                              void* d_out, int out_size, void* d_ws, size_t ws_size,
                              hipStream_t stream) {
  const float*      x   = (const float*)d_in[0];
  const long long*  ei  = (const long long*)d_in[1];
  const float* W1  = (const float*)d_in[2];
  const float* as1 = (const float*)d_in[3];
  const float* ad1 = (const float*)d_in[4];
  const float* b1  = (const float*)d_in[5];
  const float* W2  = (const float*)d_in[6];
  const float* as2 = (const float*)d_in[7];
  const float* ad2 = (const float*)d_in[8];
  const float* b2  = (const float*)d_in[9];
  const float* W3  = (const float*)d_in[10];
  const float* as3 = (const float*)d_in[11];
  const float* ad3 = (const float*)d_in[12];
  const float* b3  = (const float*)d_in[13];

  const int N = in_sizes[0] / 128;
  const int E = in_sizes[1] / 2;
  const int Etot = E + N;

  // Workspace carve-up (fp32), 256B-aligned blocks
  float* ws = (float*)d_ws;
  size_t off = 0;
  auto carve = [&](size_t nelem) -> float* {
    float* p = ws + off;
    off += (nelem + 63) & ~(size_t)63;
    return p;
  };
  float* xcur  = carve((size_t)N * 128);
  float* hbuf  = carve((size_t)N * 128);
  float* accum = carve((size_t)N * 128);
  float* al_s  = carve((size_t)N * 2);
  float* al_d  = carve((size_t)N * 2);
  float* mbuf  = carve((size_t)N * 2);
  float* zbuf  = carve((size_t)N * 2);
  float* exbuf = carve((size_t)Etot * 2);

  auto run_layer = [&](const float* xin, const float* Wp, const float* asp,
                       const float* adp, const float* bp,
                       int Hh, int C, float* yout, int do_relu) {
    const int OUT = Hh * C;
    hipMemsetAsync(zbuf,  0, (size_t)N * Hh  * sizeof(float), stream);
    hipMemsetAsync(accum, 0, (size_t)N * OUT * sizeof(float), stream);
    {
      long long n = (long long)N * Hh;
      fill_kernel<<<(unsigned)((n + 255) / 256), 256, 0, stream>>>(mbuf, -INFINITY, n);
    }
    dim3 gg((N + 15) / 16, OUT / 16);
    gemm_wmma_kernel<<<gg, 32, 0, stream>>>(xin, Wp, hbuf, N, 128, OUT);
    logits_kernel<<<(N + 255) / 256, 256, 0, stream>>>(hbuf, asp, adp, al_s, al_d, N, Hh, C);
    edge_max_kernel<<<(Etot + 255) / 256, 256, 0, stream>>>(ei, E, N, al_s, al_d, mbuf, Hh);
    edge_expsum_kernel<<<(Etot + 255) / 256, 256, 0, stream>>>(ei, E, N, al_s, al_d,
                                                               mbuf, zbuf, exbuf, Hh);
    const int gshift = (OUT == 128) ? 5 : 4;          // float4 groups per edge
    const long long tot = (long long)Etot << gshift;
    aggregate_kernel<<<(unsigned)((tot + 255) / 256), 256, 0, stream>>>(
        ei, E, N, hbuf, exbuf, zbuf, accum, Hh, C, gshift);
    const long long ntot = (long long)N * OUT;
    bias_act_kernel<<<(unsigned)((ntot + 255) / 256), 256, 0, stream>>>(
        accum, bp, yout, ntot, OUT - 1, do_relu);
  };

  run_layer(x,    W1, as1, ad1, b1, 2, 64, xcur,          1);
  run_layer(xcur, W2, as2, ad2, b2, 2, 64, xcur,          1);
  run_layer(xcur, W3, as3, ad3, b3, 1, 64, (float*)d_out, 0);
}